// TrainableGATLayer_86904368267968
// MI455X (gfx1250) — compile-verified
//
#include <hip/hip_runtime.h>
#include <hip/hip_bf16.h>

typedef float v2f __attribute__((ext_vector_type(2)));
typedef float v8f __attribute__((ext_vector_type(8)));

// LDS layout for weights: K-pairs stored contiguously so a B-frag is one
// aligned ds_load_b64.  pair p = k/2 lives at sW[p*PSTRIDE + col*2 + (k&1)].
// PSTRIDE = 160 floats (128 data + 32 pad) => consecutive pairs are 640B
// apart = 2.5 x 256B bank span, so the two wave halves (pair 2kk vs 2kk+1)
// hit disjoint halves of the 64 banks -> conflict-free.
#define PSTRIDE 160
#define WLDS    (32 * PSTRIDE)   // 5120 floats = 20KB per 64x64 matrix

// ---------- helpers: order-preserving float<->uint encoding for atomic max ----------
__device__ __forceinline__ unsigned enc_f32(float f) {
    unsigned u = __float_as_uint(f);
    return (u & 0x80000000u) ? ~u : (u | 0x80000000u);
}
__device__ __forceinline__ float dec_f32(unsigned e) {
    return (e & 0x80000000u) ? __uint_as_float(e ^ 0x80000000u) : __uint_as_float(~e);
}

__device__ __forceinline__ void stage_weight_pairs(float* sW, const float* __restrict__ W) {
    for (int i = threadIdx.x; i < 64 * 64; i += blockDim.x) {
        const int k = i >> 6, col = i & 63;
        sW[(k >> 1) * PSTRIDE + col * 2 + (k & 1)] = W[i];
    }
}

// =====================================================================
// Kernel A: xl = x2 @ W_l + b_l ; xr = x2 @ W_r + b_r   (WMMA f32 16x16x4)
// A-frag (16x4 f32): lanes 0-15 hold {K=0,K=1} of row M=lane,
//                    lanes 16-31 hold {K=2,K=3} of row M=lane-16.
// B-frag (4x16 f32): lanes 0-15 hold {K=0,K=1} of col N=lane,
//                    lanes 16-31 hold {K=2,K=3} of col N=lane-16.
// C/D (16x16 f32):   VGPR v: lanes 0-15 -> (M=v,N=lane); 16-31 -> (M=v+8,N=lane-16)
// =====================================================================
__global__ void node_xform_kernel(const float* __restrict__ x2,
                                  const float* __restrict__ Wl, const float* __restrict__ bl,
                                  const float* __restrict__ Wr, const float* __restrict__ br,
                                  float* __restrict__ xl, float* __restrict__ xr, int NT)
{
    __shared__ float sWl[WLDS];
    __shared__ float sWr[WLDS];
    __shared__ float sbl[64], sbr[64];
    stage_weight_pairs(sWl, Wl);
    stage_weight_pairs(sWr, Wr);
    for (int i = threadIdx.x; i < 64; i += blockDim.x) { sbl[i] = bl[i]; sbr[i] = br[i]; }
    __syncthreads();

    const int lane    = threadIdx.x & 31;
    const int wave    = threadIdx.x >> 5;
    const int wpb     = blockDim.x >> 5;
    const int l15     = lane & 15;
    const int half    = lane >> 4;             // 0 or 1
    const int numTiles = NT >> 4;              // NT = 50000, multiple of 16

    for (int tile = blockIdx.x * wpb + wave; tile < numTiles; tile += gridDim.x * wpb) {
        v8f accl[4] = {v8f{}, v8f{}, v8f{}, v8f{}};
        v8f accr[4] = {v8f{}, v8f{}, v8f{}, v8f{}};
        const float* xrow = x2 + (size_t)(tile * 16 + l15) * 64;
        #pragma unroll 4
        for (int kk = 0; kk < 16; ++kk) {
            const int kbase = kk * 4 + half * 2;           // even
            const v2f a = *(const v2f*)(xrow + kbase);     // global_load_b64
            const int pbase = (kk * 2 + half) * PSTRIDE;   // K-pair row in LDS
            #pragma unroll
            for (int nb = 0; nb < 4; ++nb) {
                const int col = nb * 16 + l15;
                const v2f bL = *(const v2f*)(sWl + pbase + col * 2);  // ds_load_b64
                const v2f bR = *(const v2f*)(sWr + pbase + col * 2);
                accl[nb] = __builtin_amdgcn_wmma_f32_16x16x4_f32(
                    false, a, false, bL, (short)0, accl[nb], false, false);
                accr[nb] = __builtin_amdgcn_wmma_f32_16x16x4_f32(
                    false, a, false, bR, (short)0, accr[nb], false, false);
            }
        }
        #pragma unroll
        for (int nb = 0; nb < 4; ++nb) {
            const int col = nb * 16 + l15;
            #pragma unroll
            for (int v = 0; v < 8; ++v) {
                const size_t row = (size_t)(tile * 16 + v + half * 8);
                xl[row * 64 + col] = accl[nb][v] + sbl[col];
                xr[row * 64 + col] = accr[nb][v] + sbr[col];
            }
        }
    }
}

// =====================================================================
// Kernel: init accumulators (must run every call; harness poisons ws once)
// =====================================================================
__global__ void init_ws_kernel(unsigned* __restrict__ smax, float* __restrict__ denom,
                               float* __restrict__ agg, int NT)
{
    const long long n = (long long)NT * 64;
    for (long long i = (long long)blockIdx.x * blockDim.x + threadIdx.x; i < n;
         i += (long long)gridDim.x * blockDim.x) {
        agg[i] = 0.0f;
        if (i < NT) { smax[i] = 0u; denom[i] = 0.0f; }   // enc(x) > 0 always; 0 == -infinity
    }
}

// =====================================================================
// Kernel B: per-edge score + segment max.  One wave32 per edge; lane owns
// channels {2*lane, 2*lane+1} -> coalesced 256B gathers, L2-resident tables.
// =====================================================================
__global__ void edge_score_kernel(const int* __restrict__ src, const int* __restrict__ dst,
                                  const float* __restrict__ eattr,
                                  const float* __restrict__ We, const float* __restrict__ att,
                                  const float* __restrict__ xl, const float* __restrict__ xr,
                                  float* __restrict__ score, unsigned* __restrict__ smax,
                                  long long E)
{
    const int lane = threadIdx.x & 31;
    const long long wid = (long long)blockIdx.x * (blockDim.x >> 5) + (threadIdx.x >> 5);
    const long long nw  = (long long)gridDim.x * (blockDim.x >> 5);
    const float2 we = ((const float2*)We)[lane];
    const float2 at = ((const float2*)att)[lane];
    for (long long e = wid; e < E; e += nw) {
        const int s = src[e];
        const int d = dst[e];
        const float ea = eattr[e];
        const float2 l2 = ((const float2*)(xl + (size_t)s * 64))[lane];
        const float2 r2 = ((const float2*)(xr + (size_t)d * 64))[lane];
        float m0 = l2.x + r2.x + ea * we.x;
        float m1 = l2.y + r2.y + ea * we.y;
        m0 = (m0 > 0.0f) ? m0 : 0.2f * m0;       // leaky_relu(., 0.2)
        m1 = (m1 > 0.0f) ? m1 : 0.2f * m1;
        float p = at.x * m0 + at.y * m1;
        #pragma unroll
        for (int off = 16; off > 0; off >>= 1) p += __shfl_xor(p, off, 32);
        if (lane == 0) {
            score[e] = p;
            atomicMax(&smax[d], enc_f32(p));     // GLOBAL_ATOMIC_MAX_U32
        }
    }
}

// =====================================================================
// Kernel C: denom[d] = sum(exp(score - smax[d]))  (thread per edge)
// =====================================================================
__global__ void edge_denom_kernel(const int* __restrict__ dst, const float* __restrict__ score,
                                  const unsigned* __restrict__ smax, float* __restrict__ denom,
                                  long long E)
{
    for (long long e = (long long)blockIdx.x * blockDim.x + threadIdx.x; e < E;
         e += (long long)gridDim.x * blockDim.x) {
        const int d = dst[e];
        const float ex = __expf(score[e] - dec_f32(smax[d]));
        atomicAdd(&denom[d], ex);                // GLOBAL_ATOMIC_ADD_F32
    }
}

// =====================================================================
// Kernel D: agg[dst] += alpha * xl[src]   (wave per edge, 2 ch / lane)
// =====================================================================
__global__ void edge_agg_kernel(const int* __restrict__ src, const int* __restrict__ dst,
                                const float* __restrict__ score, const unsigned* __restrict__ smax,
                                const float* __restrict__ denom, const float* __restrict__ xl,
                                float* __restrict__ agg, long long E)
{
    const int lane = threadIdx.x & 31;
    const long long wid = (long long)blockIdx.x * (blockDim.x >> 5) + (threadIdx.x >> 5);
    const long long nw  = (long long)gridDim.x * (blockDim.x >> 5);
    for (long long e = wid; e < E; e += nw) {
        const int s = src[e];
        const int d = dst[e];
        const float alpha = __expf(score[e] - dec_f32(smax[d])) / denom[d];
        const float2 l2 = ((const float2*)(xl + (size_t)s * 64))[lane];
        float* ap = agg + (size_t)d * 64 + 2 * lane;
        atomicAdd(ap,     alpha * l2.x);
        atomicAdd(ap + 1, alpha * l2.y);
    }
}

// =====================================================================
// Kernel E: out = (agg + bias) @ W_fc + b_fc   (WMMA f32 16x16x4)
// =====================================================================
__global__ void out_xform_kernel(const float* __restrict__ agg, const float* __restrict__ bias,
                                 const float* __restrict__ Wfc, const float* __restrict__ bfc,
                                 float* __restrict__ out, int NT)
{
    __shared__ float sW[WLDS];
    __shared__ float sb[64], sbias[64];
    stage_weight_pairs(sW, Wfc);
    for (int i = threadIdx.x; i < 64; i += blockDim.x) { sb[i] = bfc[i]; sbias[i] = bias[i]; }
    __syncthreads();

    const int lane = threadIdx.x & 31;
    const int wave = threadIdx.x >> 5;
    const int wpb  = blockDim.x >> 5;
    const int l15  = lane & 15;
    const int half = lane >> 4;
    const int numTiles = NT >> 4;

    for (int tile = blockIdx.x * wpb + wave; tile < numTiles; tile += gridDim.x * wpb) {
        v8f acc[4] = {v8f{}, v8f{}, v8f{}, v8f{}};
        const float* arow = agg + (size_t)(tile * 16 + l15) * 64;
        #pragma unroll 4
        for (int kk = 0; kk < 16; ++kk) {
            const int kbase = kk * 4 + half * 2;
            v2f a = *(const v2f*)(arow + kbase);
            a.x += sbias[kbase];
            a.y += sbias[kbase + 1];
            const int pbase = (kk * 2 + half) * PSTRIDE;
            #pragma unroll
            for (int nb = 0; nb < 4; ++nb) {
                const int col = nb * 16 + l15;
                const v2f b = *(const v2f*)(sW + pbase + col * 2);
                acc[nb] = __builtin_amdgcn_wmma_f32_16x16x4_f32(
                    false, a, false, b, (short)0, acc[nb], false, false);
            }
        }
        #pragma unroll
        for (int nb = 0; nb < 4; ++nb) {
            const int col = nb * 16 + l15;
            #pragma unroll
            for (int v = 0; v < 8; ++v) {
                const size_t row = (size_t)(tile * 16 + v + half * 8);
                out[row * 64 + col] = acc[nb][v] + sb[col];
            }
        }
    }
}

// =====================================================================
extern "C" void kernel_launch(void* const* d_in, const int* in_sizes, int n_in,
                              void* d_out, int out_size, void* d_ws, size_t ws_size,
                              hipStream_t stream) {
    const float* x      = (const float*)d_in[0];   // [NT, 64] (T==1 so contiguous)
    const int*   eidx   = (const int*)d_in[1];     // [2, E]
    const float* eattr  = (const float*)d_in[2];   // [E, 1]
    const float* W_l    = (const float*)d_in[3];
    const float* b_l    = (const float*)d_in[4];
    const float* W_r    = (const float*)d_in[5];
    const float* b_r    = (const float*)d_in[6];
    const float* W_e    = (const float*)d_in[7];   // [1, 64]
    const float* att    = (const float*)d_in[8];   // [64]
    const float* bias   = (const float*)d_in[9];   // [64]
    const float* W_fc   = (const float*)d_in[10];
    const float* b_fc   = (const float*)d_in[11];

    const int NT       = in_sizes[0] / 64;         // 50000
    const long long E  = (long long)in_sizes[2];   // 1.6M
    const int* src = eidx;
    const int* dst = eidx + E;

    // workspace layout (floats)
    float* ws     = (float*)d_ws;
    float* xl     = ws;
    float* xr     = xl + (size_t)NT * 64;
    float* agg    = xr + (size_t)NT * 64;
    float* score  = agg + (size_t)NT * 64;
    unsigned* smax = (unsigned*)(score + E);
    float* denom  = (float*)(smax + NT);

    const int numTiles = NT / 16;
    const int gemmBlocks = (numTiles + 7) / 8;          // 8 waves / block, 1 tile / wave

    node_xform_kernel<<<gemmBlocks, 256, 0, stream>>>(x, W_l, b_l, W_r, b_r, xl, xr, NT);
    init_ws_kernel<<<4096, 256, 0, stream>>>(smax, denom, agg, NT);
    edge_score_kernel<<<8192, 256, 0, stream>>>(src, dst, eattr, W_e, att, xl, xr, score, smax, E);
    edge_denom_kernel<<<8192, 256, 0, stream>>>(dst, score, smax, denom, E);
    edge_agg_kernel<<<8192, 256, 0, stream>>>(src, dst, score, smax, denom, xl, agg, E);
    out_xform_kernel<<<gemmBlocks, 256, 0, stream>>>(agg, bias, W_fc, b_fc, (float*)d_out, NT);
}